// ParallelIFS_39462159516152
// MI455X (gfx1250) — compile-verified
//
#include <hip/hip_runtime.h>
#include <cstdint>
#include <cstddef>

// ---------------------------------------------------------------------------
// ParallelIFS on MI455X (gfx1250)
//
// Strategy: associative affine-map scan (3 kernels on one stream):
//   pass1: per (chunk, batch) compose the chunk's affine map (A, c)
//   pass2: per batch, serially chain chunk maps -> start point of each chunk
//   pass3: per (chunk, batch) replay chunk from known start, stream outputs
//
// CDNA5 features: async global->LDS staging of the 56KB function tables
// (ASYNCcnt + s_wait_asynccnt), LDS gathers (ds_load_b128/b64/b32),
// non-temporal output stores. WMMA is intentionally NOT used: the math is
// per-lane 2x2 affine recurrences gathered by index -- no dense tile exists.
// ---------------------------------------------------------------------------

#define NF_MAX 2048     // table size (reference: 2048); LDS: 32K + 16K + 8K = 56KB
#define AS_GLOBAL __attribute__((address_space(1)))
#define AS_LDS    __attribute__((address_space(3)))

typedef int v4i __attribute__((vector_size(4 * sizeof(int))));
typedef int v2i __attribute__((vector_size(2 * sizeof(int))));

#if defined(__gfx1250__) && \
    __has_builtin(__builtin_amdgcn_global_load_async_to_lds_b128) && \
    __has_builtin(__builtin_amdgcn_global_load_async_to_lds_b64)
#define IFS_HAVE_ASYNC_LDS 1
#else
#define IFS_HAVE_ASYNC_LDS 0
#endif

__device__ __forceinline__ void ifs_wait_async_then_barrier() {
#if IFS_HAVE_ASYNC_LDS
#if __has_builtin(__builtin_amdgcn_s_wait_asynccnt)
    __builtin_amdgcn_s_wait_asynccnt(0);
#else
    asm volatile("s_wait_asynccnt 0" ::: "memory");
#endif
#endif
    __syncthreads();
}

// -------------------------- Pass 1: chunk compose ---------------------------
// block = one chunk (blockIdx.x = c), thread = batch b. Coalesced index reads.
__global__ __launch_bounds__(256)
void ifs_pass1_compose(const float* __restrict__ gW,   // [NF,2,2]
                       const float* __restrict__ gB,   // [NF,2]
                       const int*   __restrict__ gIdx, // [T,B]
                       float* __restrict__ chunkMaps,  // [C,B,6]
                       int T, int B, int NF, int L) {
    __shared__ __align__(16) float4 sW[NF_MAX];   // {W00,W01,W10,W11}
    __shared__ __align__(16) float2 sB[NF_MAX];   // {b0,b1}

    const float4* gW4 = reinterpret_cast<const float4*>(gW);
    const float2* gB2 = reinterpret_cast<const float2*>(gB);
    for (int i = threadIdx.x; i < NF; i += blockDim.x) {
#if IFS_HAVE_ASYNC_LDS
        __builtin_amdgcn_global_load_async_to_lds_b128(
            (AS_GLOBAL v4i*)(gW4 + i), (AS_LDS v4i*)(sW + i), 0, 0);
        __builtin_amdgcn_global_load_async_to_lds_b64(
            (AS_GLOBAL v2i*)(gB2 + i), (AS_LDS v2i*)(sB + i), 0, 0);
#else
        sW[i] = gW4[i];
        sB[i] = gB2[i];
#endif
    }
    ifs_wait_async_then_barrier();

    const int c = blockIdx.x;
    const int b = threadIdx.x;
    if (b >= B) return;

    // identity affine map
    float a00 = 1.f, a01 = 0.f, a10 = 0.f, a11 = 1.f, c0 = 0.f, c1 = 0.f;

    const int t0 = c * L;
    const int t1 = (t0 + L < T) ? (t0 + L) : T;
    for (int t = t0; t < t1; ++t) {
        const int id = gIdx[(size_t)t * B + b];     // coalesced across wave
        const float4 w  = sW[id];                   // ds_load_b128
        const float2 bb = sB[id];                   // ds_load_b64
        // (A,c) <- (W*A, W*c + b)
        const float n00 = fmaf(w.x, a00, w.y * a10);
        const float n01 = fmaf(w.x, a01, w.y * a11);
        const float n10 = fmaf(w.z, a00, w.w * a10);
        const float n11 = fmaf(w.z, a01, w.w * a11);
        const float nc0 = fmaf(w.x, c0, fmaf(w.y, c1, bb.x));
        const float nc1 = fmaf(w.z, c0, fmaf(w.w, c1, bb.y));
        a00 = n00; a01 = n01; a10 = n10; a11 = n11; c0 = nc0; c1 = nc1;
    }

    float* m = chunkMaps + ((size_t)c * B + b) * 6;
    m[0] = a00; m[1] = a01; m[2] = a10; m[3] = a11; m[4] = c0; m[5] = c1;
}

// ----------------------- Pass 2: chain chunk prefixes -----------------------
// One thread per batch: C sequential affine applies (cheap; maps are L2-hot).
__global__ __launch_bounds__(256)
void ifs_pass2_chain(const float* __restrict__ point,      // [B,2]
                     const float* __restrict__ chunkMaps,  // [C,B,6]
                     float* __restrict__ chunkStart,       // [C,B,2]
                     int B, int C) {
    const int b = blockIdx.x * blockDim.x + threadIdx.x;
    if (b >= B) return;
    float p0 = point[(size_t)b * 2 + 0];
    float p1 = point[(size_t)b * 2 + 1];
    for (int c = 0; c < C; ++c) {
        float* s = chunkStart + ((size_t)c * B + b) * 2;
        s[0] = p0; s[1] = p1;
        const float* m = chunkMaps + ((size_t)c * B + b) * 6;
        const float q0 = fmaf(m[0], p0, fmaf(m[1], p1, m[4]));
        const float q1 = fmaf(m[2], p0, fmaf(m[3], p1, m[5]));
        p0 = q0; p1 = q1;
    }
}

// ------------------------- Pass 3: replay + emit ----------------------------
__global__ __launch_bounds__(256)
void ifs_pass3_replay(const float* __restrict__ gW,        // [NF,2,2]
                      const float* __restrict__ gB,        // [NF,2]
                      const float* __restrict__ gOp,       // [NF]
                      const int*   __restrict__ gIdx,      // [T,B]
                      const float* __restrict__ chunkStart,// [C,B,2]
                      float* __restrict__ out,             // [T*B-REMOVE, 3]
                      int T, int B, int NF, int L, int REMOVE) {
    __shared__ __align__(16) float4 sW[NF_MAX];
    __shared__ __align__(16) float2 sB[NF_MAX];
    __shared__ __align__(16) float  sOp[NF_MAX];

    const float4* gW4  = reinterpret_cast<const float4*>(gW);
    const float2* gB2  = reinterpret_cast<const float2*>(gB);
    const float4* gOp4 = reinterpret_cast<const float4*>(gOp);
    const int nf4 = NF >> 2;
    for (int i = threadIdx.x; i < NF; i += blockDim.x) {
#if IFS_HAVE_ASYNC_LDS
        __builtin_amdgcn_global_load_async_to_lds_b128(
            (AS_GLOBAL v4i*)(gW4 + i), (AS_LDS v4i*)(sW + i), 0, 0);
        __builtin_amdgcn_global_load_async_to_lds_b64(
            (AS_GLOBAL v2i*)(gB2 + i), (AS_LDS v2i*)(sB + i), 0, 0);
        if (i < nf4) {
            __builtin_amdgcn_global_load_async_to_lds_b128(
                (AS_GLOBAL v4i*)(gOp4 + i), (AS_LDS v4i*)(sOp + 4 * i), 0, 0);
        }
#else
        sW[i] = gW4[i];
        sB[i] = gB2[i];
        sOp[i] = gOp[i];
#endif
    }
    ifs_wait_async_then_barrier();

    const int c = blockIdx.x;
    const int b = threadIdx.x;
    if (b >= B) return;

    float p0 = chunkStart[((size_t)c * B + b) * 2 + 0];
    float p1 = chunkStart[((size_t)c * B + b) * 2 + 1];

    const int t0 = c * L;
    const int t1 = (t0 + L < T) ? (t0 + L) : T;
    for (int t = t0; t < t1; ++t) {
        const int id = gIdx[(size_t)t * B + b];     // L2-resident (read in pass1)
        const float4 w  = sW[id];
        const float2 bb = sB[id];
        const float  op = sOp[id];
        const float q0 = fmaf(w.x, p0, fmaf(w.y, p1, bb.x));
        const float q1 = fmaf(w.z, p0, fmaf(w.w, p1, bb.y));
        p0 = q0; p1 = q1;

        const long long g = (long long)t * B + b;   // global row
        const long long r = g - REMOVE;
        if (r >= 0) {
            float* o = out + (size_t)r * 3;
            __builtin_nontemporal_store(p0, o + 0);  // no reuse -> NT, spare L2
            __builtin_nontemporal_store(p1, o + 1);
            __builtin_nontemporal_store(op, o + 2);
        }
    }
}

// ------------------------------- launcher -----------------------------------
extern "C" void kernel_launch(void* const* d_in, const int* in_sizes, int n_in,
                              void* d_out, int out_size, void* d_ws, size_t ws_size,
                              hipStream_t stream) {
    (void)n_in; (void)out_size; (void)ws_size;

    const float* point = (const float*)d_in[0];   // [B,2,1]
    const float* W     = (const float*)d_in[1];   // [NF,2,2]
    const float* Bi    = (const float*)d_in[2];   // [NF,2,1]
    const float* Op    = (const float*)d_in[3];   // [NF]
    const int*   idx   = (const int*)  d_in[4];   // [T,B]

    const int B  = in_sizes[0] / 2;
    const int NF = in_sizes[3];
    const int T  = in_sizes[4] / B;
    const int REMOVE = 10 * B;

    const int L = 100;                 // chunk length
    const int C = (T + L - 1) / L;     // chunks: B*C = 102400 threads of work

    // workspace: chunkMaps [C,B,6] floats, then chunkStart [C,B,2] floats
    float* chunkMaps  = (float*)d_ws;
    float* chunkStart = chunkMaps + (size_t)C * B * 6;

    ifs_pass1_compose<<<C, B, 0, stream>>>(W, Bi, idx, chunkMaps, T, B, NF, L);
    ifs_pass2_chain<<<(B + 255) / 256, 256, 0, stream>>>(point, chunkMaps,
                                                         chunkStart, B, C);
    ifs_pass3_replay<<<C, B, 0, stream>>>(W, Bi, Op, idx, chunkStart,
                                          (float*)d_out, T, B, NF, L, REMOVE);
}